// ParallelxLSTMClassifierInference_20899310862412
// MI455X (gfx1250) — compile-verified
//
#include <hip/hip_runtime.h>
#include <math.h>

static constexpr int NB = 4;      // batch
static constexpr int NS = 1000;   // seq
static constexpr int NFI = 12;    // input features
static constexpr int ND = 256;    // model dim
static constexpr int NHd = 8;     // heads
static constexpr int NDh = 32;    // slstm head dim
static constexpr int NIn = 512;   // mlstm inner
static constexpr int NMd = 64;    // mlstm head dim
static constexpr int NUp = 384;   // ffn hidden
static constexpr int NMe = 16;    // meta
static constexpr int NCl = 27;    // classes
static constexpr int NKc = 4;     // conv taps
static constexpr int SPAD = 1008; // padded seq for V^T
static constexpr float EPSF = 1e-5f;

typedef float v2f __attribute__((ext_vector_type(2)));
typedef float v8f __attribute__((ext_vector_type(8)));

__device__ __forceinline__ float sigmoidf_(float x){ return 1.f/(1.f+expf(-x)); }
__device__ __forceinline__ float logsigf_(float x){ return fminf(x,0.f) - log1pf(expf(-fabsf(x))); }
__device__ __forceinline__ float siluf_(float x){ return x*sigmoidf_(x); }

__device__ __forceinline__ float block_reduce_sum(float v, float* sh){
  int t = threadIdx.x;
  sh[t]=v; __syncthreads();
  for (int st = blockDim.x>>1; st>0; st>>=1){ if(t<st) sh[t]+=sh[t+st]; __syncthreads(); }
  float r = sh[0]; __syncthreads(); return r;
}

// ---------------- embedding + LN + relu ----------------
__global__ void k_embed(const float* __restrict__ x, const float* __restrict__ W,
                        const float* __restrict__ bias, const float* __restrict__ g,
                        const float* __restrict__ be, float* __restrict__ out){
  __shared__ float sh[ND];
  int row = blockIdx.x; int t = threadIdx.x;
  const float* xr = x + (size_t)row*NFI;
  float acc = bias[t];
  #pragma unroll
  for (int k=0;k<NFI;k++) acc += xr[k]*W[t*NFI+k];
  float mu = block_reduce_sum(acc, sh)*(1.f/ND);
  float dv = acc-mu;
  float var = block_reduce_sum(dv*dv, sh)*(1.f/ND);
  float y = dv*rsqrtf(var+EPSF)*g[t] + be[t];
  out[(size_t)row*ND+t] = fmaxf(y,0.f);
}

__global__ void k_layernorm(const float* __restrict__ in, const float* __restrict__ g,
                            const float* __restrict__ b, float* __restrict__ out){
  __shared__ float sh[ND];
  int row=blockIdx.x, t=threadIdx.x;
  float v = in[(size_t)row*ND+t];
  float mu = block_reduce_sum(v, sh)*(1.f/ND);
  float dv=v-mu;
  float var = block_reduce_sum(dv*dv, sh)*(1.f/ND);
  out[(size_t)row*ND+t] = dv*rsqrtf(var+EPSF)*g[t] + b[t];
}

__global__ void k_copy(const float* __restrict__ a, float* __restrict__ o, int n){
  int i=blockIdx.x*blockDim.x+threadIdx.x; if(i<n) o[i]=a[i];
}

// causal conv (K=4) + silu; blockDim == C
__global__ void k_conv_silu(const float* __restrict__ in, const float* __restrict__ w,
                            const float* __restrict__ cb, float* __restrict__ out,
                            int C, int in_stride){
  int b=blockIdx.x/NS, s=blockIdx.x%NS, c=threadIdx.x;
  float acc = cb[c];
  #pragma unroll
  for (int k=0;k<NKc;k++){
    int sp = s-(NKc-1)+k;
    if (sp>=0) acc += in[(size_t)(b*NS+sp)*in_stride + c]*w[c*NKc+k];
  }
  out[(size_t)(b*NS+s)*C + c] = siluf_(acc);
}

// sLSTM per-head 32x32 projections -> wx[(s*B+b), h, g, o]
__global__ void k_headwise_s(const float* __restrict__ xc, const float* __restrict__ xl,
    const float* __restrict__ Wi, const float* __restrict__ Wf,
    const float* __restrict__ Wz, const float* __restrict__ Wo,
    float* __restrict__ wx){
  int row=blockIdx.x; int b=row/NS, s=row%NS;
  int t=threadIdx.x; int h=t>>5, o=t&31;
  const float* Ws[4]={Wi,Wf,Wz,Wo};
  float* wxr = wx + ((size_t)(s*NB+b)*NHd + h)*4*NDh;
  for (int g=0; g<4; g++){
    const float* src = (g<2)? xc : xl;
    const float* sv = src + (size_t)row*ND + h*NDh;
    const float* wv = Ws[g] + (size_t)(h*NDh + o)*NDh;
    float acc=0.f;
    #pragma unroll
    for (int d=0; d<NDh; d++) acc += wv[d]*sv[d];
    wxr[g*NDh+o]=acc;
  }
}

// sLSTM recurrence: one block per (b,h), 128 threads (g=tid/32, o=tid%32)
__global__ void k_slstm_scan(const float* __restrict__ wx, const float* __restrict__ R,
    const float* __restrict__ bvec, float* __restrict__ ys){
  int bh=blockIdx.x; int b=bh>>3, h=bh&7;
  int t=threadIdx.x; int g=t>>5, o=t&31;
  float r[NDh];
  const float* Rp = R + (size_t)((h*4+g)*NDh + o)*NDh;
  #pragma unroll
  for (int d=0;d<NDh;d++) r[d]=Rp[d];
  float bias = bvec[(h*4+g)*NDh+o];
  __shared__ float ysh[NDh];
  __shared__ float raw[4][NDh];
  float c=0.f, n=0.f, mm=0.f;
  if (t<NDh) ysh[t]=0.f;
  __syncthreads();
  for (int ts=0; ts<NS; ts++){
    const float* wxp = wx + ((size_t)(ts*NB+b)*NHd + h)*4*NDh;
    float acc = bias + wxp[g*NDh+o];
    #pragma unroll
    for (int d=0;d<NDh;d++) acc += ysh[d]*r[d];
    raw[g][o]=acc;
    __syncthreads();
    if (t<NDh){
      float ir=raw[0][t], fr=raw[1][t], zr=raw[2][t], orr=raw[3][t];
      float lg = mm + logsigf_(fr);
      float mnew = (n==0.f)? ir : fmaxf(ir, lg);
      float i_=expf(ir-mnew), f_=expf(lg-mnew);
      float cn=f_*c + i_*tanhf(zr);
      float nn=f_*n + i_;
      float yn=sigmoidf_(orr)*cn/nn;
      c=cn; n=nn; mm=mnew;
      ysh[t]=yn;
      ys[((size_t)(b*NS+ts)*NHd + h)*NDh + t] = yn;
    }
    __syncthreads();
  }
}

// sLSTM group-norm + residual add into xs
__global__ void k_mhnorm_res(const float* __restrict__ ys, const float* __restrict__ g,
                             float* __restrict__ xs){
  __shared__ float sh[ND];
  int row=blockIdx.x, t=threadIdx.x, hb=(t>>5)<<5;
  float v = ys[(size_t)row*ND + t];
  sh[t]=v; __syncthreads();
  float mu=0.f; for(int j=0;j<NDh;j++) mu+=sh[hb+j]; mu*=(1.f/NDh);
  float var=0.f; for(int j=0;j<NDh;j++){ float d_=sh[hb+j]-mu; var+=d_*d_; } var*=(1.f/NDh);
  xs[(size_t)row*ND+t] += (v-mu)*rsqrtf(var+EPSF)*g[t];
}

// ---------------- fp32 WMMA GEMM: C[m][n] (+)= sum_k A[m][k]*W[n][k] ----------------
// block = 128 threads (4 waves); wave w -> n-tile blockIdx.y*64 + w*16
__global__ void k_gemm_wmma(const float* __restrict__ A, const float* __restrict__ W,
                            float* __restrict__ C, int M, int N, int Kd, int ACC){
  int l = threadIdx.x & 31, wv = threadIdx.x >> 5;
  int hi = l >> 4, ln = l & 15;
  int m0 = blockIdx.x*16;
  int n0 = blockIdx.y*64 + wv*16;
  const float* ar = A + (size_t)(m0+ln)*Kd + 2*hi;
  const float* wr = W + (size_t)(n0+ln)*Kd + 2*hi;
  v8f acc;
  if (ACC){
    #pragma unroll
    for (int r=0;r<8;r++) acc[r] = C[(size_t)(m0+r+8*hi)*N + n0+ln];
  } else {
    #pragma unroll
    for (int r=0;r<8;r++) acc[r]=0.f;
  }
  for (int k=0;k<Kd;k+=4){
    v2f a = *(const v2f*)(ar + k);
    v2f b = *(const v2f*)(wr + k);
    acc = __builtin_amdgcn_wmma_f32_16x16x4_f32(false, a, false, b, (short)0, acc, false, false);
  }
  #pragma unroll
  for (int r=0;r<8;r++) C[(size_t)(m0+r+8*hi)*N + n0+ln] = acc[r];
}

__global__ void k_gelu_mul(const float* __restrict__ gu, float* __restrict__ mid){
  int row=blockIdx.x, t=threadIdx.x;
  float x = gu[(size_t)row*(2*NUp)+t];
  float u = gu[(size_t)row*(2*NUp)+NUp+t];
  float ge = 0.5f*x*(1.f+erff(x*0.70710678118654752f));
  mid[(size_t)row*NUp+t] = ge*u;
}

// mLSTM q/k/v: 4x4 block-diagonal projections; blockDim = 512
__global__ void k_headwise_m(const float* __restrict__ xc, const float* __restrict__ xm,
    const float* __restrict__ Wq, const float* __restrict__ Wk, const float* __restrict__ Wv,
    float* __restrict__ q, float* __restrict__ k, float* __restrict__ v){
  int row=blockIdx.x, t=threadIdx.x;
  int hb=t>>2, o=t&3;
  const float* xcr = xc + (size_t)row*NIn + hb*4;
  const float* xmr = xm + (size_t)row*(2*NIn) + hb*4;
  float aq=0.f, ak=0.f, av=0.f;
  #pragma unroll
  for (int d=0;d<4;d++){
    aq += Wq[(hb*4+o)*4+d]*xcr[d];
    ak += Wk[(hb*4+o)*4+d]*xcr[d];
    av += Wv[(hb*4+o)*4+d]*xmr[d];
  }
  q[(size_t)row*NIn+t]=aq; k[(size_t)row*NIn+t]=ak; v[(size_t)row*NIn+t]=av;
}

// gates: ipre/fpre[b,h,s] = [q,k,v] . Wig[h] + b
__global__ void k_gates(const float* __restrict__ q, const float* __restrict__ k,
    const float* __restrict__ v,
    const float* __restrict__ Wig, const float* __restrict__ big,
    const float* __restrict__ Wfg, const float* __restrict__ bfg,
    float* __restrict__ ipre, float* __restrict__ fpre){
  __shared__ float sh[256];
  int row=blockIdx.x, b=row/NS, s=row%NS, t=threadIdx.x;
  for (int h=0;h<NHd;h++){
    float pi=0.f, pf=0.f;
    for (int idx=t; idx<NIn; idx+=256){
      float qq=q[(size_t)row*NIn+idx], kk=k[(size_t)row*NIn+idx], vv=v[(size_t)row*NIn+idx];
      pi += Wig[h*3*NIn+idx]*qq + Wig[h*3*NIn+NIn+idx]*kk + Wig[h*3*NIn+2*NIn+idx]*vv;
      pf += Wfg[h*3*NIn+idx]*qq + Wfg[h*3*NIn+NIn+idx]*kk + Wfg[h*3*NIn+2*NIn+idx]*vv;
    }
    float ti = block_reduce_sum(pi, sh);
    float tf = block_reduce_sum(pf, sh);
    if (t==0){
      ipre[(size_t)(b*NHd+h)*NS+s] = ti + big[h];
      fpre[(size_t)(b*NHd+h)*NS+s] = tf + bfg[h];
    }
  }
}

// per (b,h): C[s]=cumsum(logsig(f)); mlog[q]=C[q]+prefixmax(ipre[k]-C[k])
__global__ void k_gate_scan(const float* __restrict__ ipre, const float* __restrict__ fpre,
                            float* __restrict__ ccum, float* __restrict__ mlog){
  if (threadIdx.x!=0) return;
  int bh=blockIdx.x;
  const float* ip=ipre+(size_t)bh*NS; const float* fp=fpre+(size_t)bh*NS;
  float* cc=ccum+(size_t)bh*NS; float* ml=mlog+(size_t)bh*NS;
  float cs=0.f, rm=-1e30f;
  for (int s=0;s<NS;s++){
    cs += logsigf_(fp[s]);
    float a = ip[s]-cs;
    rm = fmaxf(rm,a);
    cc[s]=cs; ml[s]=cs+rm;
  }
}

// V^T: vt[(bh*64+d)*SPAD + s], zero padded for s in [NS,SPAD)
__global__ void k_vtrans(const float* __restrict__ v, float* __restrict__ vt){
  int rowid = blockIdx.x; // bh*64 + d
  int bh = rowid>>6, d = rowid&63;
  int b=bh>>3, h=bh&7;
  float* o = vt + (size_t)rowid*SPAD;
  for (int s=threadIdx.x; s<SPAD; s+=256)
    o[s] = (s<NS)? v[(size_t)(b*NS+s)*NIn + h*NMd + d] : 0.f;
}

// causal decay attention; 1 wave per block; grid (ceil(S/16), B*NH)
__global__ void k_attn(const float* __restrict__ qb, const float* __restrict__ kb,
    const float* __restrict__ vt, const float* __restrict__ ccum,
    const float* __restrict__ ipre, const float* __restrict__ mlog,
    float* __restrict__ hatt){
  __shared__ float stile[16][20];
  int qt=blockIdx.x, bh=blockIdx.y;
  int b=bh>>3, h=bh&7;
  int l=threadIdx.x, hi=l>>4, ln=l&15;
  const float* Qp = qb + (size_t)b*NS*NIn + h*NMd;
  const float* Kp = kb + (size_t)b*NS*NIn + h*NMd;
  const float* Vp = vt + (size_t)bh*NMd*SPAD;
  const float* CC = ccum + (size_t)bh*NS;
  const float* IP = ipre + (size_t)bh*NS;
  const float* ML = mlog + (size_t)bh*NS;

  int qrow = qt*16 + ln; int qrc = (qrow<NS)? qrow : (NS-1);
  v2f qa[16];
  #pragma unroll
  for (int kc=0;kc<16;kc++) qa[kc] = *(const v2f*)(Qp + (size_t)qrc*NIn + kc*4 + 2*hi);
  float cq[8], ml[8];
  #pragma unroll
  for (int r=0;r<8;r++){ int qq=qt*16+r+8*hi; if(qq>=NS) qq=NS-1; cq[r]=CC[qq]; ml[r]=ML[qq]; }
  v8f hacc[4];
  #pragma unroll
  for (int nt=0;nt<4;nt++){
    #pragma unroll
    for(int r=0;r<8;r++) hacc[nt][r]=0.f;
  }
  float rsum[8];
  #pragma unroll
  for (int r=0;r<8;r++) rsum[r]=0.f;

  for (int kt=0; kt<=qt; kt++){
    int k0=kt*16;
    int kn=k0+ln; int knc=(kn<NS)?kn:(NS-1);
    float ck=CC[knc], ik=IP[knc];
    v8f sacc;
    #pragma unroll
    for (int r=0;r<8;r++) sacc[r]=0.f;
    const float* Kr = Kp + (size_t)knc*NIn + 2*hi;
    #pragma unroll
    for (int kc=0;kc<16;kc++){
      v2f bb = *(const v2f*)(Kr + kc*4);
      sacc = __builtin_amdgcn_wmma_f32_16x16x4_f32(false, qa[kc], false, bb, (short)0, sacc, false, false);
    }
    #pragma unroll
    for (int r=0;r<8;r++){
      int qq=qt*16+r+8*hi;
      float v_=0.f;
      if (qq<NS && kn<=qq) v_ = sacc[r]*0.125f*expf(cq[r]-ck+ik-ml[r]);
      rsum[r]+=v_;
      stile[r+8*hi][ln]=v_;
    }
    __syncthreads();
    #pragma unroll
    for (int nt=0;nt<4;nt++){
      #pragma unroll
      for (int kc2=0;kc2<4;kc2++){
        v2f aa; aa[0]=stile[ln][kc2*4+2*hi]; aa[1]=stile[ln][kc2*4+2*hi+1];
        int d = nt*16+ln;
        v2f bb = *(const v2f*)(Vp + (size_t)d*SPAD + k0 + kc2*4 + 2*hi);
        hacc[nt] = __builtin_amdgcn_wmma_f32_16x16x4_f32(false, aa, false, bb, (short)0, hacc[nt], false, false);
      }
    }
    __syncthreads();
  }
  #pragma unroll
  for (int r=0;r<8;r++){
    float rs=rsum[r];
    rs += __shfl_xor(rs,1,32); rs += __shfl_xor(rs,2,32);
    rs += __shfl_xor(rs,4,32); rs += __shfl_xor(rs,8,32);
    float norm = fmaxf(fabsf(rs), expf(-ml[r])) + 1e-6f;
    int qq=qt*16+r+8*hi;
    if (qq<NS){
      #pragma unroll
      for (int nt=0;nt<4;nt++)
        hatt[(size_t)(b*NS+qq)*NIn + h*NMd + nt*16 + ln] = hacc[nt][r]/norm;
    }
  }
}

// mLSTM head-norm + skip + silu(z) gate; blockDim = 512
__global__ void k_mhn_gate(const float* __restrict__ hatt, const float* __restrict__ g,
    const float* __restrict__ skip, const float* __restrict__ xc,
    const float* __restrict__ up, float* __restrict__ out){
  __shared__ float sh[NIn];
  int row=blockIdx.x, t=threadIdx.x, hb=(t>>6)<<6;
  float v = hatt[(size_t)row*NIn+t];
  sh[t]=v; __syncthreads();
  float mu=0.f; for(int j=0;j<NMd;j++) mu+=sh[hb+j]; mu*=(1.f/NMd);
  float var=0.f; for(int j=0;j<NMd;j++){ float d_=sh[hb+j]-mu; var+=d_*d_; } var*=(1.f/NMd);
  float hn = (v-mu)*rsqrtf(var+EPSF)*g[t];
  float z = up[(size_t)row*(2*NIn) + NIn + t];
  out[(size_t)row*NIn+t] = (hn + skip[t]*xc[(size_t)row*NIn+t]) * siluf_(z);
}

__global__ void k_pool(const float* __restrict__ a, const float* __restrict__ bb,
                       float* __restrict__ pool){
  __shared__ float ssum[128]; __shared__ float smax[128];
  int b=blockIdx.x/ND, d=blockIdx.x%ND, t=threadIdx.x;
  float sm=0.f, mx=-1e30f;
  for (int s=t; s<NS; s+=128){
    float f = 0.5f*(a[(size_t)(b*NS+s)*ND+d] + bb[(size_t)(b*NS+s)*ND+d]);
    sm+=f; mx=fmaxf(mx,f);
  }
  ssum[t]=sm; smax[t]=mx; __syncthreads();
  for (int st=64; st>0; st>>=1){
    if(t<st){ ssum[t]+=ssum[t+st]; smax[t]=fmaxf(smax[t],smax[t+st]); }
    __syncthreads();
  }
  if (t==0){ pool[(size_t)b*2*ND + d] = ssum[0]*(1.f/NS); pool[(size_t)b*2*ND + ND + d] = smax[0]; }
}

__global__ void k_head(const float* __restrict__ pool, const float* __restrict__ m,
    const float* __restrict__ mW, const float* __restrict__ mb,
    const float* __restrict__ c1W, const float* __restrict__ c1b,
    const float* __restrict__ c2W, const float* __restrict__ c2b,
    const float* __restrict__ c3W, const float* __restrict__ c3b,
    float* __restrict__ out){
  __shared__ float cmb[2*ND+32];
  __shared__ float c1o[256];
  __shared__ float c2o[128];
  int b=blockIdx.x, t=threadIdx.x;
  for (int i=t;i<2*ND;i+=256) cmb[i]=pool[(size_t)b*2*ND+i];
  if (t<32){
    float acc=mb[t];
    for (int k=0;k<NMe;k++) acc += mW[t*NMe+k]*m[b*NMe+k];
    cmb[2*ND+t]=fmaxf(acc,0.f);
  }
  __syncthreads();
  {
    float acc=c1b[t];
    for (int k=0;k<2*ND+32;k++) acc += c1W[t*(2*ND+32)+k]*cmb[k];
    c1o[t]=fmaxf(acc,0.f);
  }
  __syncthreads();
  if (t<128){
    float acc=c2b[t];
    for (int k=0;k<256;k++) acc += c2W[t*256+k]*c1o[k];
    c2o[t]=fmaxf(acc,0.f);
  }
  __syncthreads();
  if (t<NCl){
    float acc=c3b[t];
    for (int k=0;k<128;k++) acc += c3W[t*128+k]*c2o[k];
    out[b*NCl+t] = sigmoidf_(acc);
  }
}

extern "C" void kernel_launch(void* const* d_in, const int* in_sizes, int n_in,
                              void* d_out, int out_size, void* d_ws, size_t ws_size,
                              hipStream_t stream){
  (void)in_sizes; (void)n_in; (void)out_size; (void)ws_size;
  const float* x = (const float*)d_in[0];
  const float* m = (const float*)d_in[1];
  #define PP(i) ((const float*)d_in[(i)])
  const int P_EMB_W=2,P_EMB_B=3,P_ELN_G=4,P_ELN_B=5,P_SLN_G=6,P_SLN_B=7,P_MLN_G=8,P_MLN_B=9;
  const int SB0=10, MB0=40;
  const int P_META_W=70,P_META_B=71,P_C1W=72,P_C1B=73,P_C2W=74,P_C2B=75,P_C3W=76,P_C3B=77;

  float* ws=(float*)d_ws;
  const size_t O_H=0, O_XS=1024000, O_XMB=2048000, A0=3072000;
  // sLSTM phase scratch
  const size_t A_LN=A0, A_XC=A0+1024000, A_WX=A0+2048000, A_YS=A0+6144000,
               A_GU=A0+7168000, A_MID=A0+10240000;
  // mLSTM phase scratch (aliases sLSTM scratch)
  const size_t B_LN=A0, B_UP2=A0+1024000, B_XCM=A0+5120000, B_Q=A0+7168000,
               B_K=A0+9216000, B_V=A0+11264000, B_VT=A0+13312000, B_HATT=A0+15392000,
               B_GD=A0+17440000, B_IPRE=A0+19488000, B_FPRE=A0+19520000,
               B_CCUM=A0+19552000, B_MLOG=A0+19584000;

  const int rows = NB*NS; // 4000

  k_embed<<<rows,256,0,stream>>>(x, PP(P_EMB_W), PP(P_EMB_B), PP(P_ELN_G), PP(P_ELN_B), ws+O_H);
  k_copy<<<(rows*ND+255)/256,256,0,stream>>>(ws+O_H, ws+O_XS, rows*ND);
  k_copy<<<(rows*ND+255)/256,256,0,stream>>>(ws+O_H, ws+O_XMB, rows*ND);

  for (int sb=0; sb<2; sb++){
    int pb = SB0 + sb*15;
    const float *LN1G=PP(pb+0),*LN1B=PP(pb+1),*CW=PP(pb+2),*CB=PP(pb+3),*WI=PP(pb+4),
      *WF=PP(pb+5),*WZ=PP(pb+6),*WO=PP(pb+7),*RR=PP(pb+8),*BV=PP(pb+9),*GNG=PP(pb+10),
      *LN2G=PP(pb+11),*LN2B=PP(pb+12),*WUP=PP(pb+13),*WDN=PP(pb+14);
    k_layernorm<<<rows,256,0,stream>>>(ws+O_XS, LN1G, LN1B, ws+A_LN);
    k_conv_silu<<<rows,256,0,stream>>>(ws+A_LN, CW, CB, ws+A_XC, ND, ND);
    k_headwise_s<<<rows,256,0,stream>>>(ws+A_XC, ws+A_LN, WI,WF,WZ,WO, ws+A_WX);
    k_slstm_scan<<<NB*NHd,128,0,stream>>>(ws+A_WX, RR, BV, ws+A_YS);
    k_mhnorm_res<<<rows,256,0,stream>>>(ws+A_YS, GNG, ws+O_XS);
    k_layernorm<<<rows,256,0,stream>>>(ws+O_XS, LN2G, LN2B, ws+A_LN);
    k_gemm_wmma<<<dim3(rows/16,(2*NUp)/64),128,0,stream>>>(ws+A_LN, WUP, ws+A_GU, rows, 2*NUp, ND, 0);
    k_gelu_mul<<<rows,NUp,0,stream>>>(ws+A_GU, ws+A_MID);
    k_gemm_wmma<<<dim3(rows/16,ND/64),128,0,stream>>>(ws+A_MID, WDN, ws+O_XS, rows, ND, NUp, 1);
  }

  for (int mb2=0; mb2<2; mb2++){
    int pb = MB0 + mb2*15;
    const float *LNG=PP(pb+0),*LNB=PP(pb+1),*WUPm=PP(pb+2),*CW=PP(pb+3),*CB=PP(pb+4),
      *WQ=PP(pb+5),*WK2=PP(pb+6),*WV=PP(pb+7),*WIG=PP(pb+8),*BIG=PP(pb+9),*WFG=PP(pb+10),
      *BFG=PP(pb+11),*MHG=PP(pb+12),*SKP=PP(pb+13),*WDN=PP(pb+14);
    k_layernorm<<<rows,256,0,stream>>>(ws+O_XMB, LNG, LNB, ws+B_LN);
    k_gemm_wmma<<<dim3(rows/16,(2*NIn)/64),128,0,stream>>>(ws+B_LN, WUPm, ws+B_UP2, rows, 2*NIn, ND, 0);
    k_conv_silu<<<rows,NIn,0,stream>>>(ws+B_UP2, CW, CB, ws+B_XCM, NIn, 2*NIn);
    k_headwise_m<<<rows,NIn,0,stream>>>(ws+B_XCM, ws+B_UP2, WQ,WK2,WV, ws+B_Q, ws+B_K, ws+B_V);
    k_gates<<<rows,256,0,stream>>>(ws+B_Q, ws+B_K, ws+B_V, WIG,BIG,WFG,BFG, ws+B_IPRE, ws+B_FPRE);
    k_gate_scan<<<NB*NHd,32,0,stream>>>(ws+B_IPRE, ws+B_FPRE, ws+B_CCUM, ws+B_MLOG);
    k_vtrans<<<NB*NHd*NMd,256,0,stream>>>(ws+B_V, ws+B_VT);
    k_attn<<<dim3((NS+15)/16, NB*NHd),32,0,stream>>>(ws+B_Q, ws+B_K, ws+B_VT,
                                                     ws+B_CCUM, ws+B_IPRE, ws+B_MLOG, ws+B_HATT);
    k_mhn_gate<<<rows,NIn,0,stream>>>(ws+B_HATT, MHG, SKP, ws+B_XCM, ws+B_UP2, ws+B_GD);
    k_gemm_wmma<<<dim3(rows/16,ND/64),128,0,stream>>>(ws+B_GD, WDN, ws+O_XMB, rows, ND, NIn, 1);
  }

  const size_t C_XSLN=A0, C_XMBLN=A0+1024000, C_POOL=A0+2048000;
  k_layernorm<<<rows,256,0,stream>>>(ws+O_XS, PP(P_SLN_G), PP(P_SLN_B), ws+C_XSLN);
  k_layernorm<<<rows,256,0,stream>>>(ws+O_XMB, PP(P_MLN_G), PP(P_MLN_B), ws+C_XMBLN);
  k_pool<<<NB*ND,128,0,stream>>>(ws+C_XSLN, ws+C_XMBLN, ws+C_POOL);
  k_head<<<NB,256,0,stream>>>(ws+C_POOL, m, PP(P_META_W),PP(P_META_B),
                              PP(P_C1W),PP(P_C1B),PP(P_C2W),PP(P_C2B),PP(P_C3W),PP(P_C3B),
                              (float*)d_out);
  #undef PP
}